// MarkovGCNR_29592324669623
// MI455X (gfx1250) — compile-verified
//
#include <hip/hip_runtime.h>
#include <hip/hip_bf16.h>
#include <math.h>

// MarkovGCNR on MI455X (gfx1250, wave32).
//
// x0  = GCN(features, W0, b0, edges0, ew=1)
// x1  = 0.5*GCN(relu(x0), W1, b1, edges1, ew1) + 0.5*x0
// out = log_softmax(GCN(relu(x1), W2, b2, edges2, ew2))
// GCN(x)[c] = sum_e{col=c} dinv[row]*ew*dinv[col]*(xW)[row] + dinv[c]^2*(xW)[c] + b
//
// Roofline: GEMMs are ~11 GFLOP (trivial); the edge scatter dominates
// (~3.3 GB of gather+atomic traffic, L2-resident since each h is 51 MB < 192 MB L2).
// GEMMs use fp32-exact V_WMMA_F32_16X16X4_F32; scatters use scalar (SGPR) edge
// metadata + float4 vector payload + global_atomic_add_f32 + global_prefetch.

typedef __attribute__((ext_vector_type(2))) float v2f;
typedef __attribute__((ext_vector_type(8))) float v8f;

#define ALPHA_C 0.5f

// ---------------------------------------------------------------- utilities
__global__ void fill_f32(float* __restrict__ p, float v, int n) {
  int i = blockIdx.x * blockDim.x + threadIdx.x;
  if (i < n) p[i] = v;
}

// deg[col] += ew  (edges are int64 in the reference -> long long)
__global__ void accum_deg(const long long* __restrict__ ed, const float* __restrict__ ew,
                          float* __restrict__ deg, int ne) {
  int e = blockIdx.x * blockDim.x + threadIdx.x;
  if (e >= ne) return;
  int c = (int)ed[(size_t)ne + e];
  float w = ew ? ew[e] : 1.0f;
  atomicAdd(&deg[c], w);
}

// deg -> dinv in place (deg >= 1 always due to the self loop init)
__global__ void rsqrt_inplace(float* __restrict__ d, int n) {
  int i = blockIdx.x * blockDim.x + threadIdx.x;
  if (i < n) d[i] = rsqrtf(d[i]);
}

// ---------------------------------------------------------------- WMMA GEMM
// H[n x NOUT] = (RELU_IN ? relu(X) : X)[n x K] @ W[K x NOUT]
// One 16-node tile per block; wave w computes the 16x16 tile at columns
// [16w, 16w+16). blockDim.x = 32 * ceil(NOUT/16).
//
// V_WMMA_F32_16X16X4_F32 fragments (ISA 7.12.2, wave32):
//   A (16x4):  vgpr j = K=j (lanes 0-15) / K=j+2 (lanes 16-31), M = lane&15
//   B (4x16):  symmetric; N = lane&15
//   C/D:       vgpr i = row i (lanes 0-15) / row i+8 (lanes 16-31)
template<int K, int NOUT, bool RELU_IN>
__global__ void gemm_wmma_f32(const float* __restrict__ X, const float* __restrict__ W,
                              float* __restrict__ H, int n) {
  const int lane = threadIdx.x & 31;
  const int wave = threadIdx.x >> 5;
  const int hlf  = lane >> 4;
  const int l16  = lane & 15;
  const int m0   = blockIdx.x * 16;
  const int c0   = wave * 16;

  const int arow = (m0 + l16 < n) ? (m0 + l16) : (n - 1);        // clamp, EXEC stays full
  const int bcol = (c0 + l16 < NOUT) ? (c0 + l16) : (NOUT - 1);  // clamp, mask on store

  const float* __restrict__ xr = X + (size_t)arow * K;
  v8f acc0 = {}, acc1 = {};                    // two chains -> more WMMA/load overlap
  for (int k = 0; k < K; k += 8) {
    float a0x = xr[k + 2 * hlf], a0y = xr[k + 2 * hlf + 1];
    float a1x = xr[k + 4 + 2 * hlf], a1y = xr[k + 4 + 2 * hlf + 1];
    if (RELU_IN) {
      a0x = fmaxf(a0x, 0.0f); a0y = fmaxf(a0y, 0.0f);
      a1x = fmaxf(a1x, 0.0f); a1y = fmaxf(a1y, 0.0f);
    }
    v2f a0; a0.x = a0x; a0.y = a0y;
    v2f a1; a1.x = a1x; a1.y = a1y;
    v2f b0; b0.x = W[(size_t)(k + 2 * hlf) * NOUT + bcol];
            b0.y = W[(size_t)(k + 2 * hlf + 1) * NOUT + bcol];
    v2f b1; b1.x = W[(size_t)(k + 4 + 2 * hlf) * NOUT + bcol];
            b1.y = W[(size_t)(k + 4 + 2 * hlf + 1) * NOUT + bcol];
    acc0 = __builtin_amdgcn_wmma_f32_16x16x4_f32(false, a0, false, b0, (short)0, acc0,
                                                 false, false);
    acc1 = __builtin_amdgcn_wmma_f32_16x16x4_f32(false, a1, false, b1, (short)0, acc1,
                                                 false, false);
  }
  v8f acc = acc0 + acc1;

  const int col = c0 + l16;
  if (m0 + 16 <= n) {                          // uniform fast path: full row tile
#pragma unroll
    for (int i = 0; i < 8; ++i) {
      if (NOUT % 16 == 0 || col < NOUT)        // constexpr-true for NOUT=128
        H[(size_t)(m0 + i + 8 * hlf) * NOUT + col] = acc[i];
    }
  } else {                                      // tail tile (absent for N%16==0)
#pragma unroll
    for (int i = 0; i < 8; ++i) {
      int row = m0 + i + 8 * hlf;
      if (row < n && col < NOUT)
        H[(size_t)row * NOUT + col] = acc[i];
    }
  }
}

// ---------------------------------------------------------------- scatter
// 4 edges per wave. Edge metadata (row, col, ew, dinv) is wave-uniform ->
// force through readfirstlane so it lowers to scalar (s_load) traffic; the
// vector lanes carry only the float4 payload + 4 global_atomic_add_f32.
// Prefetch all 4 gathered h rows up front to overlap L2 latency.
__global__ void scatter128(const long long* __restrict__ ed, const float* __restrict__ ew,
                           const float* __restrict__ dinv, const float* __restrict__ h,
                           float* __restrict__ out, int ne) {
  const int lane = threadIdx.x & 31;
  int wv = (int)(((size_t)blockIdx.x * blockDim.x + threadIdx.x) >> 5);
  const int base = __builtin_amdgcn_readfirstlane(wv) * 4;
  if (base >= ne) return;                      // wave-uniform
  const int cnt = (ne - base < 4) ? (ne - base) : 4;

  int rr[4], cc[4];
  float nm[4];
#pragma unroll
  for (int j = 0; j < 4; ++j) {
    if (j < cnt) {
      int r = (int)ed[base + j];
      int c = (int)ed[(size_t)ne + base + j];
      float w = ew ? ew[base + j] : 1.0f;
      rr[j] = r; cc[j] = c;
      nm[j] = dinv[r] * w * dinv[c];
      __builtin_prefetch(h + (size_t)r * 128 + lane * 4, 0, 1);  // global_prefetch
    }
  }
#pragma unroll
  for (int j = 0; j < 4; ++j) {
    if (j < cnt) {
      const float4* hr = (const float4*)(h + (size_t)rr[j] * 128);
      float4 v = hr[lane];
      float nrm = nm[j];
      float* oc = out + (size_t)cc[j] * 128 + lane * 4;
      atomicAdd(oc + 0, nrm * v.x);
      atomicAdd(oc + 1, nrm * v.y);
      atomicAdd(oc + 2, nrm * v.z);
      atomicAdd(oc + 3, nrm * v.w);
    }
  }
}

// One wave per edge, 40 feats: lane handles f=lane, lanes 0-7 also f=lane+32.
__global__ void scatter40(const long long* __restrict__ ed, const float* __restrict__ ew,
                          const float* __restrict__ dinv, const float* __restrict__ h,
                          float* __restrict__ out, int ne) {
  const int lane = threadIdx.x & 31;
  int wv = (int)(((size_t)blockIdx.x * blockDim.x + threadIdx.x) >> 5);
  const int wid = __builtin_amdgcn_readfirstlane(wv);
  if (wid >= ne) return;
  int r = (int)ed[wid];
  int c = (int)ed[(size_t)ne + wid];
  float nrm = dinv[r] * ew[wid] * dinv[c];
  const float* hr = h + (size_t)r * 40;
  float* oc = out + (size_t)c * 40;
  atomicAdd(oc + lane, nrm * hr[lane]);
  if (lane < 8) atomicAdd(oc + 32 + lane, nrm * hr[32 + lane]);
}

// ---------------------------------------------------------------- finalizers
// float4-vectorized over [N, 128]: thread i handles node i>>5, feats 4*(i&31).
// x0 += dinv^2 * h + b
__global__ void finalize_l0(float4* __restrict__ x0, const float4* __restrict__ h,
                            const float* __restrict__ dinv, const float4* __restrict__ b,
                            int n) {
  int i = blockIdx.x * blockDim.x + threadIdx.x;
  if (i >= n * 32) return;
  int node = i >> 5, f4 = i & 31;
  float dd = dinv[node]; dd *= dd;
  float4 x = x0[i], hv = h[i], bv = b[f4];
  x.x += dd * hv.x + bv.x; x.y += dd * hv.y + bv.y;
  x.z += dd * hv.z + bv.z; x.w += dd * hv.w + bv.w;
  x0[i] = x;
}

// x1 = alpha*(agg1 + dinv^2*h + b) + (1-alpha)*x0   (agg1 already in x1)
__global__ void finalize_l1(float4* __restrict__ x1, const float4* __restrict__ h,
                            const float* __restrict__ dinv, const float4* __restrict__ b,
                            const float4* __restrict__ x0, int n) {
  int i = blockIdx.x * blockDim.x + threadIdx.x;
  if (i >= n * 32) return;
  int node = i >> 5, f4 = i & 31;
  float dd = dinv[node]; dd *= dd;
  float4 x = x1[i], hv = h[i], bv = b[f4], p = x0[i];
  x.x = ALPHA_C * (x.x + dd * hv.x + bv.x) + (1.0f - ALPHA_C) * p.x;
  x.y = ALPHA_C * (x.y + dd * hv.y + bv.y) + (1.0f - ALPHA_C) * p.y;
  x.z = ALPHA_C * (x.z + dd * hv.z + bv.z) + (1.0f - ALPHA_C) * p.z;
  x.w = ALPHA_C * (x.w + dd * hv.w + bv.w) + (1.0f - ALPHA_C) * p.w;
  x1[i] = x;
}

// out = log_softmax(agg2 + dinv^2*h2 + b2); one wave per node (40 cols).
__global__ void out_logsoftmax(float* __restrict__ out, const float* __restrict__ h2,
                               const float* __restrict__ dinv, const float* __restrict__ b2,
                               int n) {
  int node = (int)(((size_t)blockIdx.x * blockDim.x + threadIdx.x) >> 5);
  int lane = threadIdx.x & 31;
  if (node >= n) return;
  float di = dinv[node], dd = di * di;
  const float* hr = h2 + (size_t)node * 40;
  float* orow = out + (size_t)node * 40;

  float a = orow[lane] + dd * hr[lane] + b2[lane];            // lane < 32 < 40
  float bv = -INFINITY;
  if (lane < 8) bv = orow[32 + lane] + dd * hr[32 + lane] + b2[32 + lane];

  float m = fmaxf(a, bv);
  for (int off = 16; off; off >>= 1) m = fmaxf(m, __shfl_xor(m, off, 32));
  float s = expf(a - m) + (lane < 8 ? expf(bv - m) : 0.0f);
  for (int off = 16; off; off >>= 1) s += __shfl_xor(s, off, 32);
  float ls = logf(s);

  orow[lane] = a - m - ls;
  if (lane < 8) orow[32 + lane] = bv - m - ls;
}

// ---------------------------------------------------------------- driver
extern "C" void kernel_launch(void* const* d_in, const int* in_sizes, int n_in,
                              void* d_out, int out_size, void* d_ws, size_t ws_size,
                              hipStream_t stream) {
  const float*     features = (const float*)d_in[0];
  const long long* edges0   = (const long long*)d_in[1];   // int64 in reference
  const long long* edges1   = (const long long*)d_in[2];
  const long long* edges2   = (const long long*)d_in[3];
  const float*     ew1      = (const float*)d_in[4];
  const float*     ew2      = (const float*)d_in[5];
  const float*     W0       = (const float*)d_in[6];
  const float*     b0       = (const float*)d_in[7];
  const float*     W1       = (const float*)d_in[8];
  const float*     b1       = (const float*)d_in[9];
  const float*     W2       = (const float*)d_in[10];
  const float*     b2       = (const float*)d_in[11];
  float* out = (float*)d_out;

  const int n  = in_sizes[0] / 256;   // N nodes (100000, multiple of 16)
  const int E0 = in_sizes[1] / 2;
  const int E1 = in_sizes[2] / 2;
  const int E2 = in_sizes[3] / 2;

  // workspace: h[N*128] | x0[N*128] | x1[N*128] | deg/dinv[N]  (~154 MB)
  float* ws    = (float*)d_ws;
  size_t nf128 = (size_t)n * 128;
  float* bufH  = ws;
  float* bufX0 = bufH + nf128;
  float* bufX1 = bufX0 + nf128;
  float* deg   = bufX1 + nf128;

  const int T = 256;
  const int gN    = (n + T - 1) / T;
  const int gNF   = (int)((nf128 + T - 1) / T);
  const int gNF4  = ((n * 32) + T - 1) / T;            // float4 elementwise
  const int gN40  = ((n * 40) + T - 1) / T;
  const int gTile = (n + 15) / 16;
  const int gSc128_0 = (E0 + 31) / 32;                 // 4 edges/wave, 8 waves/block
  const int gSc128_1 = (E1 + 31) / 32;
  const int gSc40    = (E2 + 7) / 8;                   // 1 edge/wave

  // ---------------- layer 0 (edge weights = 1) ----------------
  fill_f32<<<gN, T, 0, stream>>>(deg, 1.0f, n);
  accum_deg<<<(E0 + T - 1) / T, T, 0, stream>>>(edges0, (const float*)nullptr, deg, E0);
  rsqrt_inplace<<<gN, T, 0, stream>>>(deg, n);
  gemm_wmma_f32<256, 128, false><<<gTile, 256, 0, stream>>>(features, W0, bufH, n);
  fill_f32<<<gNF, T, 0, stream>>>(bufX0, 0.0f, (int)nf128);
  scatter128<<<gSc128_0, T, 0, stream>>>(edges0, (const float*)nullptr, deg, bufH, bufX0, E0);
  finalize_l0<<<gNF4, T, 0, stream>>>((float4*)bufX0, (const float4*)bufH, deg,
                                      (const float4*)b0, n);

  // ---------------- layer 1 (ew1, residual blend) ----------------
  fill_f32<<<gN, T, 0, stream>>>(deg, 1.0f, n);
  accum_deg<<<(E1 + T - 1) / T, T, 0, stream>>>(edges1, ew1, deg, E1);
  rsqrt_inplace<<<gN, T, 0, stream>>>(deg, n);
  gemm_wmma_f32<128, 128, true><<<gTile, 256, 0, stream>>>(bufX0, W1, bufH, n);
  fill_f32<<<gNF, T, 0, stream>>>(bufX1, 0.0f, (int)nf128);
  scatter128<<<gSc128_1, T, 0, stream>>>(edges1, ew1, deg, bufH, bufX1, E1);
  finalize_l1<<<gNF4, T, 0, stream>>>((float4*)bufX1, (const float4*)bufH, deg,
                                      (const float4*)b1, (const float4*)bufX0, n);

  // ---------------- layer 2 (ew2, 40 outputs, log_softmax) ----------------
  fill_f32<<<gN, T, 0, stream>>>(deg, 1.0f, n);
  accum_deg<<<(E2 + T - 1) / T, T, 0, stream>>>(edges2, ew2, deg, E2);
  rsqrt_inplace<<<gN, T, 0, stream>>>(deg, n);
  gemm_wmma_f32<128, 40, true><<<gTile, 96, 0, stream>>>(bufX1, W2, bufH, n);  // 3 waves
  fill_f32<<<gN40, T, 0, stream>>>(out, 0.0f, n * 40);
  scatter40<<<gSc40, T, 0, stream>>>(edges2, ew2, deg, bufH, out, E2);
  out_logsoftmax<<<(n + 7) / 8, T, 0, stream>>>(out, bufH, deg, b2, n);
}